// Polyrigid_27865747817165
// MI455X (gfx1250) — compile-verified
//
#include <hip/hip_runtime.h>

typedef float v2f __attribute__((ext_vector_type(2)));
typedef float v8f __attribute__((ext_vector_type(8)));

// Problem constants (fixed by the reference).
constexpr int Dd = 80, Hh = 112, Ww = 112, Kk = 8;
constexpr int HW = Hh * Ww;
constexpr int NV = Dd * Hh * Ww;            // 1,003,520 = 3920 * 256 exactly
constexpr int LDS_STRIDE = 12;              // floats; 48B row -> 16B aligned

__global__ __launch_bounds__(256) void polyrigid_kernel(
    const float* __restrict__ img,   // (D,H,W)
    const float* __restrict__ rot,   // (K,9)
    const float* __restrict__ trs,   // (K,3)
    const float* __restrict__ wts,   // (D,H,W,K)
    float* __restrict__ out)         // (D,H,W)
{
  __shared__ __align__(16) float s_lept[256 * LDS_STRIDE]; // 12 KB

  const int tid  = threadIdx.x;
  const int ln   = tid & 31;          // wave32 lane
  const int wave = tid >> 5;
  const int n    = ln & 15;           // D column (matrix entry) / A row
  const int hf   = ln >> 4;           // lane half selects K pair
  const int blockBase = blockIdx.x * 256;

  // ---- Stage 1: lept = weights @ log_t via V_WMMA_F32_16X16X4_F32 ----
  // B fragment (log_t, 8x16): v0 holds row 2*hf, v1 row 2*hf+1 (chunk0); +4 (chunk1).
  auto logt = [&](int k, int t) -> float {
    if (t < 9)  return rot[k * 9 + t];
    if (t < 12) return trs[k * 3 + (t - 9)];
    return 0.0f;
  };
  v2f B0, B1;
  B0.x = logt(2 * hf + 0, n);
  B0.y = logt(2 * hf + 1, n);
  B1.x = logt(2 * hf + 4, n);
  B1.y = logt(2 * hf + 5, n);

#pragma unroll
  for (int t = 0; t < 2; ++t) {
    const int tileBase = blockBase + wave * 32 + t * 16;
    // A fragment: lane row m = n; v0/v1 = K pair (2*hf, 2*hf+1); chunk1 +4.
    const int rowOff = (tileBase + n) * Kk;
    v2f A0 = *(const v2f*)(wts + rowOff + 2 * hf);
    v2f A1 = *(const v2f*)(wts + rowOff + 2 * hf + 4);

    v8f c = {};
    c = __builtin_amdgcn_wmma_f32_16x16x4_f32(false, A0, false, B0,
                                              (short)0, c, false, false);
    c = __builtin_amdgcn_wmma_f32_16x16x4_f32(false, A1, false, B1,
                                              (short)0, c, false, false);

    // D layout: lane n holds entry n for voxels (r + 8*hf) in VGPR r.
    // Entries 12..15 are identically zero (log_t row 3 == 0) -> store only n<12.
    const int vLocalBase = wave * 32 + t * 16 + hf * 8;
    if (n < 12) {
#pragma unroll
      for (int r = 0; r < 8; ++r) {
        s_lept[(vLocalBase + r) * LDS_STRIDE + n] = c[r];
      }
    }
  }

  __syncthreads();

  // ---- Stage 2: per-lane voxel pipeline (voxel = blockBase + tid) ----
  // L = top 3x4 of the 4x4 log matrix; bottom row is exactly zero.
  float L[12];
#pragma unroll
  for (int i = 0; i < 12; ++i) L[i] = s_lept[tid * LDS_STRIDE + i];

  // expm(L) by Horner-form Taylor, order 9 (||L|| ~ 0.2 -> error << fp32 eps).
  // All iterates are affine: implicit bottom row [0,0,0,1]. Seed P = I + L/9.
  float P[12];
#pragma unroll
  for (int i = 0; i < 12; ++i) {
    const float idv = (i == 0 || i == 5 || i == 10) ? 1.0f : 0.0f;
    P[i] = fmaf(L[i], 1.0f / 9.0f, idv);
  }
#pragma unroll
  for (int j = 8; j >= 1; --j) {
    float M[12];
#pragma unroll
    for (int i = 0; i < 3; ++i) {
#pragma unroll
      for (int c = 0; c < 4; ++c) {
        // (L @ P)[i][c]; P's implicit row3 contributes L[i][3] to column 3.
        float acc = (c == 3) ? L[i * 4 + 3] : 0.0f;
        acc = fmaf(L[i * 4 + 0], P[0 * 4 + c], acc);
        acc = fmaf(L[i * 4 + 1], P[1 * 4 + c], acc);
        acc = fmaf(L[i * 4 + 2], P[2 * 4 + c], acc);
        M[i * 4 + c] = acc;
      }
    }
    const float inv = 1.0f / (float)j;
#pragma unroll
    for (int i = 0; i < 12; ++i) {
      const float idv = (i == 0 || i == 5 || i == 10) ? 1.0f : 0.0f;
      P[i] = fmaf(M[i], inv, idv);
    }
  }

  // T = view @ E (rows 0..2; row 3 = [0,0,0,1] exactly, so p3 == 1 -> no divide).
  // view = [[c,0,1,0],[0,1,0,0],[1,0,c,0],[0,0,0,1]], c = cos(-pi/2)
  const float vc = 6.1232339957367660e-17f;
  float T[12];
#pragma unroll
  for (int c = 0; c < 4; ++c) {
    T[0 * 4 + c] = fmaf(vc, P[0 * 4 + c], P[2 * 4 + c]);
    T[1 * 4 + c] = P[1 * 4 + c];
    T[2 * 4 + c] = fmaf(vc, P[2 * 4 + c], P[0 * 4 + c]);
  }

  const int v   = blockBase + tid;
  const int z   = v / HW;
  const int rem = v - z * HW;
  const int y   = rem / Ww;
  const int x   = rem - y * Ww;
  const float zc = fmaf(2.0f / (float)(Dd - 1), (float)z, -1.0f);
  const float yc = fmaf(2.0f / (float)(Hh - 1), (float)y, -1.0f);
  const float xc = fmaf(2.0f / (float)(Ww - 1), (float)x, -1.0f);

  // g = T @ [z,y,x,1] (homogeneous w == 1 exactly)
  const float gx = fmaf(T[0], zc, fmaf(T[1], yc, fmaf(T[2],  xc, T[3])));
  const float gy = fmaf(T[4], zc, fmaf(T[5], yc, fmaf(T[6],  xc, T[7])));
  const float gz = fmaf(T[8], zc, fmaf(T[9], yc, fmaf(T[10], xc, T[11])));

  // ((g+1)*S - 1)*0.5 = g*S/2 + (S-1)/2
  const float ixf = fmaf(gx, 0.5f * (float)Ww, 0.5f * (float)(Ww - 1));
  const float iyf = fmaf(gy, 0.5f * (float)Hh, 0.5f * (float)(Hh - 1));
  const float izf = fmaf(gz, 0.5f * (float)Dd, 0.5f * (float)(Dd - 1));

  const float fx0 = floorf(ixf), fy0 = floorf(iyf), fz0 = floorf(izf);
  const float fx = ixf - fx0, fy = iyf - fy0, fz = izf - fz0;
  const int x0 = (int)fx0, y0 = (int)fy0, z0 = (int)fz0;

  float acc = 0.0f;
#pragma unroll
  for (int dz = 0; dz < 2; ++dz) {
    const int   zi = z0 + dz;
    const float wz = dz ? fz : (1.0f - fz);
    const bool  zv = (zi >= 0) && (zi < Dd);
    const int   zq = min(max(zi, 0), Dd - 1);
#pragma unroll
    for (int dy = 0; dy < 2; ++dy) {
      const int   yi = y0 + dy;
      const float wy = dy ? fy : (1.0f - fy);
      const bool  yv = (yi >= 0) && (yi < Hh);
      const int   yq = min(max(yi, 0), Hh - 1);
#pragma unroll
      for (int dx = 0; dx < 2; ++dx) {
        const int   xi = x0 + dx;
        const float wx = dx ? fx : (1.0f - fx);
        const bool  xv = (xi >= 0) && (xi < Ww);
        const int   xq = min(max(xi, 0), Ww - 1);
        const float wgt = wz * wy * wx * ((zv && yv && xv) ? 1.0f : 0.0f);
        acc = fmaf(wgt, img[(zq * Hh + yq) * Ww + xq], acc);
      }
    }
  }
  out[v] = acc;
}

extern "C" void kernel_launch(void* const* d_in, const int* in_sizes, int n_in,
                              void* d_out, int out_size, void* d_ws, size_t ws_size,
                              hipStream_t stream) {
  const float* img = (const float*)d_in[0];  // (1,1,D,H,W)
  const float* rot = (const float*)d_in[1];  // (K,9)
  const float* trs = (const float*)d_in[2];  // (K,3)
  const float* wts = (const float*)d_in[3];  // (D,H,W,K)
  float* out = (float*)d_out;                // (1,1,D,H,W)

  (void)in_sizes; (void)n_in; (void)out_size; (void)d_ws; (void)ws_size;

  polyrigid_kernel<<<dim3(NV / 256), dim3(256), 0, stream>>>(img, rot, trs, wts, out);
}